// PartSelectionModule_58832462020964
// MI455X (gfx1250) — compile-verified
//
#include <hip/hip_runtime.h>

#define B_  64
#define H_  12
#define S_  577
#define P_  576            // patches = S-1
#define D_  768
#define SS_ (577 * 577)    // 332929, stride between (b,h) attention matrices

typedef __attribute__((ext_vector_type(2))) float v2f;
typedef __attribute__((ext_vector_type(8))) float v8f;

__global__ __launch_bounds__(256) void topk_pool_kernel(
    const float* __restrict__ attn,    // [B, H, S, S]
    const float* __restrict__ tokens,  // [B, P, D]
    const int*   __restrict__ topk,    // scalar
    float*       __restrict__ out)     // [B, D]
{
    __shared__ float vals[P_];                 // per-patch head-summed CLS attention
    __shared__ unsigned long long red[256];    // argmax reduction tree
    __shared__ int sel[32];                    // selected patch indices

    const int tid = threadIdx.x;
    const int b   = blockIdx.x;

    int K = *topk;
    K = (K < 1) ? 1 : ((K > 32) ? 32 : K);

    // ---------------- Phase A: sums over heads via fp32 WMMA ----------------
    // sums[p0..p0+15] = W(16 patches x 4 heads) * ones(4x16), accumulated over
    // 3 head-chunks with V_WMMA_F32_16X16X4_F32. Monotonic w.r.t. the mean, so
    // the top-k set is unchanged by skipping the /H.
#if __has_builtin(__builtin_amdgcn_wmma_f32_16x16x4_f32)
    {
        const int wave = tid >> 5;
        const int lane = tid & 31;
        const int m    = lane & 15;          // A-matrix row (patch within tile)
        const int kk   = (lane >> 4) << 1;   // A-matrix K pair: 0 or 2

        v2f ones; ones.x = 1.0f; ones.y = 1.0f;   // B = all-ones (layout-agnostic)

        for (int tile = wave; tile < P_ / 16; tile += 8) {   // 36 tiles / 8 waves
            const int p0 = tile * 16;
            v8f c = {0.f, 0.f, 0.f, 0.f, 0.f, 0.f, 0.f, 0.f};
            for (int hc = 0; hc < H_; hc += 4) {
                // A layout (16x4 f32): lanes 0-15 hold K=0,1; lanes 16-31 hold K=2,3
                v2f a;
                const size_t base =
                    ((size_t)b * H_ + (size_t)(hc + kk)) * (size_t)SS_ + 1 + p0 + m;
                a.x = attn[base];
                a.y = attn[base + SS_];
                c = __builtin_amdgcn_wmma_f32_16x16x4_f32(
                        /*neg_a=*/false, a, /*neg_b=*/false, ones,
                        /*c_mod=*/(short)0, c, /*reuse_a=*/false, /*reuse_b=*/false);
            }
            // All 16 columns of D are identical (B==ones). C/D layout:
            // lanes 0-15: VGPR v -> M=v ; lanes 16-31: VGPR v -> M=v+8.
            if (m == 0) {
                const int mb = (lane >> 4) * 8;
#pragma unroll
                for (int v = 0; v < 8; ++v) vals[p0 + mb + v] = c[v];
            }
        }
    }
#else
    for (int p = tid; p < P_; p += 256) {
        float s = 0.0f;
        for (int h = 0; h < H_; ++h)
            s += attn[((size_t)b * H_ + h) * (size_t)SS_ + 1 + p];
        vals[p] = s;
    }
#endif
    __syncthreads();

    // ---------------- Phase B: iterated argmax top-K --------------------------
    // Sums are >= 0, so IEEE bits compare monotonically as unsigned. Key packs
    // value bits high, (0xFFFFFFFF - index) low => ties break to smaller index.
    for (int k = 0; k < K; ++k) {
        unsigned long long best = 0ull;
        for (int p = tid; p < P_; p += 256) {
            const float v = vals[p];
            if (v >= 0.0f) {
                const unsigned long long key =
                    ((unsigned long long)__float_as_uint(v) << 32) |
                    (unsigned long long)(0xFFFFFFFFu - (unsigned)p);
                if (key > best) best = key;
            }
        }
        red[tid] = best;
        __syncthreads();
        for (int s = 128; s > 0; s >>= 1) {
            if (tid < s) {
                const unsigned long long o = red[tid + s];
                if (o > red[tid]) red[tid] = o;
            }
            __syncthreads();
        }
        if (tid == 0) {
            const int p = (int)(0xFFFFFFFFu - (unsigned)(red[0] & 0xFFFFFFFFull));
            sel[k] = p;
            vals[p] = -1.0f;   // remove winner
        }
        __syncthreads();
    }

    // ---------------- Phase C: gather selected rows, mean-pool ----------------
    if (tid < D_ / 4) {
        const float4* tok4 = (const float4*)tokens;
        float4 acc = make_float4(0.f, 0.f, 0.f, 0.f);
        for (int k = 0; k < K; ++k) {
            const size_t row = ((size_t)b * P_ + (size_t)sel[k]) * (D_ / 4) + tid;
            const float4 t = tok4[row];
            acc.x += t.x; acc.y += t.y; acc.z += t.z; acc.w += t.w;
        }
        const float inv = 1.0f / (float)K;
        float4 r;
        r.x = acc.x * inv; r.y = acc.y * inv; r.z = acc.z * inv; r.w = acc.w * inv;
        ((float4*)out)[(size_t)b * (D_ / 4) + tid] = r;
    }
}

extern "C" void kernel_launch(void* const* d_in, const int* in_sizes, int n_in,
                              void* d_out, int out_size, void* d_ws, size_t ws_size,
                              hipStream_t stream) {
    const float* attn   = (const float*)d_in[0];
    const float* tokens = (const float*)d_in[1];
    const int*   topk   = (const int*)d_in[2];
    float*       out    = (float*)d_out;

    topk_pool_kernel<<<B_, 256, 0, stream>>>(attn, tokens, topk, out);
}